// TestLoss1_73504070304259
// MI455X (gfx1250) — compile-verified
//
#include <hip/hip_runtime.h>
#include <math.h>

// Problem constants (match reference)
#define BB 16
#define SS 4096
#define CC 512
#define AA 0.5f

// ---- workspace layout (bytes) ----
#define CNT_OFF   0u
#define N_OFF     16777216u
#define EQLSE_OFF 16809984u
#define COLS_OFF  16842752u
#define EQPRD_OFF 16875520u
#define M_OFF     16908288u
#define SLSE_OFF  16941056u
#define WS_TOTAL  16941120u                 // multiple of 16

typedef float v2f __attribute__((ext_vector_type(2)));
typedef float v8f __attribute__((ext_vector_type(8)));

// ---- VALU-only cross-lane primitives (no ds_bpermute round trips) ----
// DPP controls: quad_perm xor1 = 0xB1, quad_perm xor2 = 0x4E,
// row_half_mirror = 0x141 (xor7 within 8), row_mirror = 0x140 (xor15 within 16)
template <int CTRL>
__device__ __forceinline__ int dpp_mov_i(int x) {
  return __builtin_amdgcn_update_dpp(0, x, CTRL, 0xF, 0xF, true);
}
template <int CTRL>
__device__ __forceinline__ float dpp_mov_f(float x) {
  return __int_as_float(dpp_mov_i<CTRL>(__float_as_int(x)));
}
// swap 16-lane halves (lanes within each half are identical when we call this)
__device__ __forceinline__ int permx16_i(int x) {
  return __builtin_amdgcn_permlanex16(x, x, 0x76543210, 0xFEDCBA98, false, false);
}
__device__ __forceinline__ float permx16_f(float x) {
  return __int_as_float(permx16_i(__float_as_int(x)));
}

// Wave32 sum reduction: 4 DPP adds fold each 16-lane half, then one
// V_WMMA_F32_16X16X4_F32 with all-ones A performs the cross-half pairing:
// D[m][n] = B[.. n] + B[.. n+16] = halfsum_lo + halfsum_hi in every lane.
// Requires EXEC all-ones (uniform control flow).
__device__ __forceinline__ float wave_reduce_sum_wmma(float x) {
  x += dpp_mov_f<0xB1>(x);
  x += dpp_mov_f<0x4E>(x);
  x += dpp_mov_f<0x141>(x);
  x += dpp_mov_f<0x140>(x);   // each 16-lane half now holds its half-sum
  v2f a; a[0] = 1.0f; a[1] = 1.0f;
  v2f b; b[0] = x;    b[1] = 0.0f;
  v8f c = {};
  c = __builtin_amdgcn_wmma_f32_16x16x4_f32(false, a, false, b, (short)0, c,
                                            false, false);
  return c[0];
}

// Wave32 argmax (max value, min index on ties) — DPP within halves,
// permlanex16 across halves. All VALU.
__device__ __forceinline__ void wave_reduce_argmax_f(float& mv, int& mi) {
#define AMAX_STEP_F(CTRL)                                                   \
  { float ov = dpp_mov_f<CTRL>(mv); int oi = dpp_mov_i<CTRL>(mi);           \
    if (ov > mv || (ov == mv && oi < mi)) { mv = ov; mi = oi; } }
  AMAX_STEP_F(0xB1)
  AMAX_STEP_F(0x4E)
  AMAX_STEP_F(0x141)
  AMAX_STEP_F(0x140)
#undef AMAX_STEP_F
  { float ov = permx16_f(mv); int oi = permx16_i(mi);
    if (ov > mv || (ov == mv && oi < mi)) { mv = ov; mi = oi; } }
}

__device__ __forceinline__ void wave_reduce_argmax_i(int& bv, int& bi) {
#define AMAX_STEP_I(CTRL)                                                   \
  { int ov = dpp_mov_i<CTRL>(bv); int oi = dpp_mov_i<CTRL>(bi);             \
    if (ov > bv || (ov == bv && oi < bi)) { bv = ov; bi = oi; } }
  AMAX_STEP_I(0xB1)
  AMAX_STEP_I(0x4E)
  AMAX_STEP_I(0x141)
  AMAX_STEP_I(0x140)
#undef AMAX_STEP_I
  { int ov = permx16_i(bv); int oi = permx16_i(bi);
    if (ov > bv || (ov == bv && oi < bi)) { bv = ov; bi = oi; } }
}

// ---------------- Pass 0: zero workspace ----------------
__global__ void zero_ws(float4* __restrict__ p, int n4) {
  int i = blockIdx.x * blockDim.x + threadIdx.x;
  int stride = gridDim.x * blockDim.x;
  float4 z = make_float4(0.f, 0.f, 0.f, 0.f);
  for (; i < n4; i += stride) p[i] = z;
}

// ---------------- Pass 1: stream predicted once ----------------
// grid = (B, S/128), block = 256 (8 waves). Each wave handles 16 rows.
__global__ void __launch_bounds__(256)
pass1(const float* __restrict__ P, const int* __restrict__ T,
      int* __restrict__ counts, int* __restrict__ n_arr,
      float* __restrict__ eq_lse, float* __restrict__ colsum,
      float* __restrict__ sum_lse) {
  const int b     = blockIdx.x;
  const int chunk = blockIdx.y;
  const int wave  = threadIdx.x >> 5;
  const int lane  = threadIdx.x & 31;
  const int s0    = chunk * 128 + wave * 16;

  float cs[16];
#pragma unroll
  for (int k = 0; k < 16; ++k) cs[k] = 0.f;
  float lse_acc = 0.f;  // same value in every lane of the wave

  for (int r = 0; r < 16; ++r) {
    const int s = s0 + r;
    const float4* row4 =
        reinterpret_cast<const float4*>(P + ((size_t)b * SS + s) * CC);
    float4 q0 = row4[lane];
    float4 q1 = row4[lane + 32];
    float4 q2 = row4[lane + 64];
    float4 q3 = row4[lane + 96];
    float v[16] = {q0.x, q0.y, q0.z, q0.w, q1.x, q1.y, q1.z, q1.w,
                   q2.x, q2.y, q2.z, q2.w, q3.x, q3.y, q3.z, q3.w};
    // lane-local argmax; columns of v[k] are (k/4)*128 + 4*lane + (k%4),
    // strictly increasing in k, so '>' keeps the first occurrence.
    float mv = v[0];
    int   mi = 4 * lane;
#pragma unroll
    for (int k = 1; k < 16; ++k) {
      int col = (k >> 2) * 128 + 4 * lane + (k & 3);
      if (v[k] > mv) { mv = v[k]; mi = col; }
    }
    wave_reduce_argmax_f(mv, mi);  // all lanes: (row max, first index)

    // sum of exp(v - max); inputs in [-~12, 0] so fast __expf is safe
    float se = 0.f;
#pragma unroll
    for (int k = 0; k < 16; ++k) se += __expf(v[k] - mv);
    se = wave_reduce_sum_wmma(se);
    float lse = mv + logf(se);
    lse_acc += lse;

    if (lane == 0) {
      int t = T[b * SS + s];
      atomicAdd(&counts[((size_t)b * CC + t) * CC + mi], 1);
      atomicAdd(&n_arr[b * CC + t], 1);
      atomicAdd(&eq_lse[b * CC + t], lse);
    }
#pragma unroll
    for (int k = 0; k < 16; ++k) cs[k] += v[k];
  }
  if (lane == 0) atomicAdd(&sum_lse[b], lse_acc);

  // block-level colsum reduce: LDS, then one global atomic per column
  __shared__ float cls[CC];
  for (int i = threadIdx.x; i < CC; i += blockDim.x) cls[i] = 0.f;
  __syncthreads();
#pragma unroll
  for (int k = 0; k < 16; ++k) {
    int col = (k >> 2) * 128 + 4 * lane + (k & 3);
    atomicAdd(&cls[col], cs[k]);
  }
  __syncthreads();
  for (int i = threadIdx.x; i < CC; i += blockDim.x)
    atomicAdd(&colsum[b * CC + i], cls[i]);
}

// ---------------- Pass 2: m[b,t] = argmax_a counts[b,t,a] ----------------
__global__ void __launch_bounds__(256)
pass2(const int* __restrict__ counts, int* __restrict__ m_arr) {
  const int row  = blockIdx.x * 8 + (threadIdx.x >> 5);  // [0, B*C)
  const int lane = threadIdx.x & 31;
  const int* crow = counts + (size_t)row * CC;
  int bv = -1, bi = CC;
  for (int a = lane; a < CC; a += 32) {
    int v = crow[a];
    if (v > bv || (v == bv && a < bi)) { bv = v; bi = a; }
  }
  wave_reduce_argmax_i(bv, bi);
  if (lane == 0) m_arr[row] = bi;
}

// ---------------- Pass 3: eq_pred[b,t] += P[b,s,m[b,t]] ----------------
__global__ void __launch_bounds__(256)
pass3(const float* __restrict__ P, const int* __restrict__ T,
      const int* __restrict__ m_arr, float* __restrict__ eq_pred) {
  int idx = blockIdx.x * blockDim.x + threadIdx.x;  // [0, B*S)
  int b = idx >> 12;  // /SS
  int t = T[idx];
  int mv = m_arr[b * CC + t];
  float p = P[(size_t)idx * CC + mv];
  atomicAdd(&eq_pred[b * CC + t], p);
}

// ---------------- Pass 4: final scalar ----------------
__global__ void __launch_bounds__(256)
pass4(const int* __restrict__ n_arr, const float* __restrict__ eq_lse,
      const float* __restrict__ eq_pred, const float* __restrict__ colsum,
      const float* __restrict__ sum_lse, const int* __restrict__ m_arr,
      float* __restrict__ out) {
  float tot = 0.f, cnt = 0.f;
  for (int i = threadIdx.x; i < BB * CC; i += 256) {  // 32 uniform iterations
    int b = i >> 9;  // /CC
    int nv = n_arr[i];
    bool present = nv > 0;
    float eq_sum   = eq_lse[i] - eq_pred[i];
    float tot_term = sum_lse[b] - colsum[b * CC + m_arr[i]];
    float ne_sum   = tot_term - eq_sum;
    float nf = (float)nv;
    float eq_loss = eq_sum / fmaxf(nf, 1.f);
    float ne_mean = ne_sum / fmaxf((float)SS - nf, 1.f);
    float denom   = (present && ne_mean != 0.f) ? ne_mean : 1.f;
    float ne_loss = 1.f / denom;
    float val = eq_loss * (1.f - AA) + AA * ne_loss;
    bool keep = present && (val != 0.f);
    tot += keep ? val : 0.f;
    cnt += keep ? 1.f : 0.f;
  }
  tot = wave_reduce_sum_wmma(tot);
  cnt = wave_reduce_sum_wmma(cnt);
  __shared__ float st[8], sc[8];
  if ((threadIdx.x & 31) == 0) {
    st[threadIdx.x >> 5] = tot;
    sc[threadIdx.x >> 5] = cnt;
  }
  __syncthreads();
  if (threadIdx.x == 0) {
    float Tt = 0.f, Ct = 0.f;
    for (int w = 0; w < 8; ++w) { Tt += st[w]; Ct += sc[w]; }
    out[0] = Tt / fmaxf(Ct, 1.f);
  }
}

extern "C" void kernel_launch(void* const* d_in, const int* in_sizes, int n_in,
                              void* d_out, int out_size, void* d_ws,
                              size_t ws_size, hipStream_t stream) {
  const float* P = (const float*)d_in[0];   // predicted (B,S,C) fp32
  const int*   T = (const int*)d_in[1];     // target (B,S) int
  float* out = (float*)d_out;

  char* ws = (char*)d_ws;
  int*   counts  = (int*)(ws + CNT_OFF);
  int*   n_arr   = (int*)(ws + N_OFF);
  float* eq_lse  = (float*)(ws + EQLSE_OFF);
  float* colsum  = (float*)(ws + COLS_OFF);
  float* eq_pred = (float*)(ws + EQPRD_OFF);
  int*   m_arr   = (int*)(ws + M_OFF);
  float* sum_lse = (float*)(ws + SLSE_OFF);

  // Pass 0: zero all accumulators (harness does not re-zero between replays)
  zero_ws<<<1024, 256, 0, stream>>>((float4*)ws, (int)(WS_TOTAL / 16));

  // Pass 1: single streaming pass over predicted (134 MB, bandwidth bound)
  dim3 g1(BB, SS / 128);
  pass1<<<g1, 256, 0, stream>>>(P, T, counts, n_arr, eq_lse, colsum, sum_lse);

  // Pass 2: per-(b,t) argmax over counts
  pass2<<<(BB * CC) / 8, 256, 0, stream>>>(counts, m_arr);

  // Pass 3: gather predicted[b,s,m[b,target]] (L2-resident) into eq_pred
  pass3<<<(BB * SS) / 256, 256, 0, stream>>>(P, T, m_arr, eq_pred);

  // Pass 4: final scalar reduction
  pass4<<<1, 256, 0, stream>>>(n_arr, eq_lse, eq_pred, colsum, sum_lse, m_arr,
                               out);
}